// STAGNNAggregator_30236569764345
// MI455X (gfx1250) — compile-verified
//
#include <hip/hip_runtime.h>
#include <math.h>

typedef __attribute__((ext_vector_type(2))) float v2f;
typedef __attribute__((ext_vector_type(8))) float v8f;

#define T_    8
#define N_    20000
#define DIN_  128
#define D_    128
#define HEADS 4
#define DH_   32
#define WW_   4
#define E_    320000
#define DPE_  8

static constexpr int   M_ROWS = T_ * N_;                 // 160000 (divisible by 16)
static constexpr float SCALE_ = 0.17677669529663687f;    // 32^-0.5

// ---------------------------------------------------------------------------
// Ordered-uint mapping so unsigned atomicMax implements float max.
// ---------------------------------------------------------------------------
__device__ __forceinline__ unsigned mapf(float x) {
  unsigned u = __float_as_uint(x);
  return (u & 0x80000000u) ? ~u : (u | 0x80000000u);
}
__device__ __forceinline__ float unmapf(unsigned u) {
  unsigned b = (u & 0x80000000u) ? (u & 0x7fffffffu) : ~u;
  return __uint_as_float(b);
}

// ---------------------------------------------------------------------------
// Fused Q/K/V projection:  out[m,n] = sum_k X[m,k] * W[n,k]  via
// V_WMMA_F32_16X16X4_F32.  One workgroup = 8 waves = one 16-row block of X
// staged once in LDS; wave w owns column tile [16w,16w+16) for all 3 weights.
//
// Fragment layouts (ISA 7.12.2):
//   A 16x4 : lanes 0-15 hold M=lane, K={k0,k0+1}; lanes 16-31 hold K={k0+2,k0+3}
//   B 4x16 : lane holds B[k0+2*half + {0,1}][n], n = lane%16  (= W[n][k...])
//   C/D    : VGPR r -> row r + 8*(lane/16), col = lane%16
// ---------------------------------------------------------------------------
__global__ __launch_bounds__(256)
void proj_wmma_kernel(const float* __restrict__ Ht,
                      const float* __restrict__ Wq,
                      const float* __restrict__ Wk,
                      const float* __restrict__ Wv,
                      float* __restrict__ Qb,
                      float* __restrict__ Kb,
                      float* __restrict__ Vb) {
  __shared__ float As[16 * 128];

  const int tid = threadIdx.x;
  const int r0  = blockIdx.x * 16;   // first global row of this block

  // Stage 16x128 contiguous tile of H_tilde into LDS (2048 floats, 256 thr).
  const float4* gsrc = (const float4*)(Ht + (size_t)r0 * 128);
  float4*       ldst = (float4*)As;
  ldst[tid]       = gsrc[tid];
  ldst[tid + 256] = gsrc[tid + 256];
  __syncthreads();

  const int lane    = tid & 31;
  const int wv      = tid >> 5;       // wave id = column tile 0..7
  const int half    = lane >> 4;
  const int lm      = lane & 15;
  const int colbase = wv * 16;

  const float* Ws[3] = {Wq, Wk, Wv};
  float*       Os[3] = {Qb, Kb, Vb};

  for (int m = 0; m < 3; ++m) {
    const float* brow = Ws[m] + (size_t)(colbase + lm) * 128;  // W row n
    __builtin_prefetch(brow, 0, 1);
    v8f acc = {0.f, 0.f, 0.f, 0.f, 0.f, 0.f, 0.f, 0.f};
#pragma unroll
    for (int kk = 0; kk < 32; ++kk) {
      const int k0 = kk * 4 + 2 * half;
      v2f a = *(const v2f*)&As[lm * 128 + k0];
      v2f b = *(const v2f*)&brow[k0];
      acc = __builtin_amdgcn_wmma_f32_16x16x4_f32(
          /*neg_a=*/false, a, /*neg_b=*/false, b,
          /*c_mod=*/(short)0, acc, /*reuse_a=*/false, /*reuse_b=*/false);
    }
    float* out = Os[m];
#pragma unroll
    for (int r = 0; r < 8; ++r) {
      const int row = r0 + r + 8 * half;
      out[(size_t)row * 128 + colbase + lm] = acc[r];
    }
  }
}

// ---------------------------------------------------------------------------
// gate = log(clip(S,0,1) + 1e-6)  and  pe_k[dt][j] = sum_p pe[dt][p]*Wkpe[j][p]
// ---------------------------------------------------------------------------
__global__ __launch_bounds__(256)
void gate_pe_kernel(const float* __restrict__ S,
                    const float* __restrict__ pe_table,
                    const float* __restrict__ Wkpe,
                    float* __restrict__ gate,
                    float* __restrict__ pek) {
  const int idx = blockIdx.x * blockDim.x + threadIdx.x;
  if (idx < T_ * N_) {
    float s = S[idx];
    s = fminf(fmaxf(s, 0.f), 1.f);
    gate[idx] = __logf(s + 1e-6f);
  }
  if (idx < (WW_ + 1) * D_) {
    const int dt = idx / D_;
    const int j  = idx - dt * D_;
    float acc = 0.f;
#pragma unroll
    for (int p = 0; p < DPE_; ++p)
      acc += pe_table[dt * DPE_ + p] * Wkpe[j * DPE_ + p];
    pek[idx] = acc;
  }
}

// ---------------------------------------------------------------------------
// Per-t init: zero output slice, denom, mapped max (0u == ordered -inf).
// ---------------------------------------------------------------------------
__global__ __launch_bounds__(256)
void init_t_kernel(float* __restrict__ outT,
                   float* __restrict__ denom,
                   unsigned* __restrict__ mxu) {
  const int idx = blockIdx.x * blockDim.x + threadIdx.x;
  if (idx < N_ * D_) outT[idx] = 0.f;
  if (idx < N_ * HEADS) { denom[idx] = 0.f; mxu[idx] = 0u; }
}

// ---------------------------------------------------------------------------
// Pass 1: per (edge, head) scores for dt = 0..min(W,t); store + atomic max.
// ---------------------------------------------------------------------------
__global__ __launch_bounds__(256)
void score_kernel(const int* __restrict__ src, const int* __restrict__ dst,
                  const float* __restrict__ Qb, const float* __restrict__ Kb,
                  const float* __restrict__ gate, const float* __restrict__ pek,
                  const float* __restrict__ rel_bias,
                  float* __restrict__ sbuf, unsigned* __restrict__ mxu, int t) {
  const int gid = blockIdx.x * blockDim.x + threadIdx.x;
  if (gid >= E_ * HEADS) return;
  const int e = gid >> 2;
  const int h = gid & 3;
  const int sN = src[e];
  const int dN = dst[e];

  const float* q = Qb + ((size_t)t * N_ + dN) * D_ + h * DH_;
  float4 q4[8];
#pragma unroll
  for (int g = 0; g < 8; ++g) q4[g] = ((const float4*)q)[g];

  const int dmax = (t < WW_) ? t : WW_;
  float best = -1e30f;
  for (int dt = 0; dt <= dmax; ++dt) {
    const float* kp = Kb + ((size_t)(t - dt) * N_ + sN) * D_ + h * DH_;
    const float* pp = pek + dt * D_ + h * DH_;
    float dot = 0.f;
#pragma unroll
    for (int g = 0; g < 8; ++g) {
      const float4 kv = ((const float4*)kp)[g];
      const float4 pv = ((const float4*)pp)[g];
      dot += q4[g].x * (kv.x + pv.x) + q4[g].y * (kv.y + pv.y) +
             q4[g].z * (kv.z + pv.z) + q4[g].w * (kv.w + pv.w);
    }
    const float sc = dot * SCALE_ + rel_bias[dt] + gate[(size_t)(t - dt) * N_ + sN];
    sbuf[((size_t)e * (WW_ + 1) + dt) * HEADS + h] = sc;
    best = fmaxf(best, sc);
  }
  atomicMax(&mxu[dN * HEADS + h], mapf(best));
}

// ---------------------------------------------------------------------------
// Pass 2: exp / segment sums.  denom += sum_dt ex;  out[dst] += ex*V[src].
// ---------------------------------------------------------------------------
__global__ __launch_bounds__(256)
void accum_kernel(const int* __restrict__ src, const int* __restrict__ dst,
                  const float* __restrict__ Vb, const float* __restrict__ sbuf,
                  const unsigned* __restrict__ mxu,
                  float* __restrict__ denom, float* __restrict__ outT, int t) {
  const int gid = blockIdx.x * blockDim.x + threadIdx.x;
  if (gid >= E_ * HEADS) return;
  const int e = gid >> 2;
  const int h = gid & 3;
  const int sN = src[e];
  const int dN = dst[e];

  const float mx   = unmapf(mxu[dN * HEADS + h]);
  const int   dmax = (t < WW_) ? t : WW_;

  float ex[WW_ + 1];
  float dsum = 0.f;
  for (int dt = 0; dt <= dmax; ++dt) {
    const float v = __expf(sbuf[((size_t)e * (WW_ + 1) + dt) * HEADS + h] - mx);
    ex[dt] = v;
    dsum += v;
  }
  atomicAdd(&denom[dN * HEADS + h], dsum);

  float* ob = outT + (size_t)dN * D_ + h * DH_;
#pragma unroll
  for (int g = 0; g < 8; ++g) {
    float ax = 0.f, ay = 0.f, az = 0.f, aw = 0.f;
    for (int dt = 0; dt <= dmax; ++dt) {
      const float4 vv = *(const float4*)(Vb + ((size_t)(t - dt) * N_ + sN) * D_ + h * DH_ + g * 4);
      ax += ex[dt] * vv.x; ay += ex[dt] * vv.y;
      az += ex[dt] * vv.z; aw += ex[dt] * vv.w;
    }
    atomicAdd(&ob[g * 4 + 0], ax);
    atomicAdd(&ob[g * 4 + 1], ay);
    atomicAdd(&ob[g * 4 + 2], az);
    atomicAdd(&ob[g * 4 + 3], aw);
  }
}

// ---------------------------------------------------------------------------
// msg = numer / max(denom, 1e-12)
// ---------------------------------------------------------------------------
__global__ __launch_bounds__(256)
void finalize_kernel(const float* __restrict__ denom, float* __restrict__ outT) {
  const int idx = blockIdx.x * blockDim.x + threadIdx.x;
  if (idx >= N_ * D_) return;
  const int n = idx >> 7;         // / 128
  const int h = (idx >> 5) & 3;   // channel/32
  const float d = denom[n * HEADS + h];
  outT[idx] = outT[idx] / fmaxf(d, 1e-12f);
}

// ---------------------------------------------------------------------------
extern "C" void kernel_launch(void* const* d_in, const int* in_sizes, int n_in,
                              void* d_out, int out_size, void* d_ws, size_t ws_size,
                              hipStream_t stream) {
  (void)in_sizes; (void)n_in; (void)out_size; (void)ws_size;

  const float* Ht       = (const float*)d_in[0];
  const float* S        = (const float*)d_in[1];
  const float* Wq       = (const float*)d_in[2];
  const float* Wkh      = (const float*)d_in[3];
  const float* Wkpe     = (const float*)d_in[4];
  const float* Wv       = (const float*)d_in[5];
  const float* pe_table = (const float*)d_in[6];
  const float* rel_bias = (const float*)d_in[7];
  const int*   src      = (const int*)d_in[8];
  const int*   dst      = (const int*)d_in[9];
  float*       out      = (float*)d_out;

  // Workspace layout (floats); every region stays 16-byte aligned.
  float* ws = (float*)d_ws;
  const size_t TND = (size_t)T_ * N_ * D_;          // 20,480,000
  float*    Qb    = ws;
  float*    Kb    = Qb + TND;
  float*    Vb    = Kb + TND;
  float*    gate  = Vb + TND;                       // T*N
  float*    pek   = gate + (size_t)T_ * N_;         // (W+1)*D
  float*    denom = pek + (size_t)(WW_ + 1) * D_;   // N*HEADS
  unsigned* mxu   = (unsigned*)(denom + (size_t)N_ * HEADS);
  float*    sbuf  = (float*)(mxu + (size_t)N_ * HEADS);  // E*(W+1)*HEADS

  // 1) Q/K/V projections (WMMA fp32 path).
  proj_wmma_kernel<<<M_ROWS / 16, 256, 0, stream>>>(Ht, Wq, Wkh, Wv, Qb, Kb, Vb);

  // 2) gate + pe_k.
  gate_pe_kernel<<<(T_ * N_ + 255) / 256, 256, 0, stream>>>(S, pe_table, Wkpe, gate, pek);

  // 3) per-timestep edge attention.
  const int EH = E_ * HEADS;
  for (int t = 0; t < T_; ++t) {
    float* outT = out + (size_t)t * N_ * D_;
    init_t_kernel<<<(N_ * D_ + 255) / 256, 256, 0, stream>>>(outT, denom, mxu);
    score_kernel<<<(EH + 255) / 256, 256, 0, stream>>>(src, dst, Qb, Kb, gate, pek,
                                                       rel_bias, sbuf, mxu, t);
    accum_kernel<<<(EH + 255) / 256, 256, 0, stream>>>(src, dst, Vb, sbuf, mxu,
                                                       denom, outT, t);
    finalize_kernel<<<(N_ * D_ + 255) / 256, 256, 0, stream>>>(denom, outT);
  }
}